// GraphRecommendationModel_49271864819740
// MI455X (gfx1250) — compile-verified
//
#include <hip/hip_runtime.h>
#include <hip/hip_bf16.h>

// Shapes from the reference
#define Bc 256
#define Lc 50
#define Nn 100   // neighbors
#define Hc 64    // hidden
#define Rrows (Bc * Lc)   // 12800

typedef __attribute__((ext_vector_type(2))) float v2f;
typedef __attribute__((ext_vector_type(8))) float v8f;
typedef __attribute__((ext_vector_type(4))) int   v4i_;

// ---- CDNA5 async global->LDS path (guarded so compile never regresses) ----
#if defined(__has_builtin)
#if __has_builtin(__builtin_amdgcn_global_load_async_to_lds_b128)
#define HAVE_ASYNC_LDS 1
#endif
#endif

#ifdef HAVE_ASYNC_LDS
typedef __attribute__((address_space(1))) v4i_ gvec_t;   // global int4
typedef __attribute__((address_space(3))) v4i_ lvec_t;   // LDS int4
// generic -> AS(1): full 64-bit flat address == global address
__device__ static inline gvec_t* to_gbl(const void* p) {
    return (gvec_t*)(unsigned long long)p;
}
// generic -> AS(3): flat-shared addr[31:0] == LDS byte offset (ISA 10.2)
__device__ static inline lvec_t* to_lds(void* p) {
    return (lvec_t*)(unsigned)(unsigned long long)p;
}
__device__ static inline void wait_async0() {
#if __has_builtin(__builtin_amdgcn_s_wait_asynccnt)
    __builtin_amdgcn_s_wait_asynccnt(0);
#else
    asm volatile("s_wait_asynccnt 0x0" ::: "memory");
#endif
}
#endif

// ---------------------------------------------------------------------------
// Kernel 1: neighbor attention streamer. One block per (b,l).
// Stages the 100x64 f32 neighbor tile ONCE into LDS (via async global->LDS
// DMA on CDNA5), computes sn[n] = relu(s0 + ne[n].wn), softmax over n,
// agg_raw = sum_n w[n]*ne[n].  Carries ~98% of all bytes -> HBM-bound.
// ---------------------------------------------------------------------------
__global__ __launch_bounds__(256) void k1_neighbor_attn(
    const int*   __restrict__ history,       // [B*L]
    const float* __restrict__ neighbor_emb,  // [B*L, N, H]
    const float* __restrict__ item_table,    // [NUM_ITEMS, H]
    const float* __restrict__ w_neigh,       // [2H]
    const float* __restrict__ b_neigh,       // [1]
    float*       __restrict__ agg_raw)       // [B*L, H]
{
    __shared__ float ne[Nn][Hc];   // 25600 B neighbor tile
    __shared__ float tv[128];      // relu scores -> exp values
    __shared__ float red[16][Hc];  // partial accumulators
    __shared__ float s_s0, s_inv;

    const int tid = threadIdx.x;
    const int row = blockIdx.x;        // b*L + l
    const int c   = tid & 15;          // column chunk: h = 4c..4c+3
    const int g   = tid >> 4;          // row group 0..15

    // phase 0: s0 = dot(hist, wa) + b_neigh   (wave 0 only)
    if (tid < 32) {
        const int hidx = history[row];
        const float* hr = item_table + (size_t)hidx * Hc;
        float p = hr[tid] * w_neigh[tid] + hr[tid + 32] * w_neigh[tid + 32];
        #pragma unroll
        for (int m = 16; m >= 1; m >>= 1) p += __shfl_xor(p, m, 32);
        if (tid == 0) s_s0 = p + b_neigh[0];
    }

    const float4 wn4 = *(const float4*)(w_neigh + Hc + 4 * c);
    const float* neb = neighbor_emb + (size_t)row * Nn * Hc;

#ifdef HAVE_ASYNC_LDS
    // phase 1a: DMA the whole tile global->LDS, no VGPR round-trip.
    // 7 passes x 16 rows x (16 lanes * 16B) ; ASYNCcnt-tracked.
    #pragma unroll
    for (int p = 0; p < 7; ++p) {
        const int n = p * 16 + g;
        if (n < Nn) {
            __builtin_amdgcn_global_load_async_to_lds_b128(
                to_gbl(neb + (size_t)n * Hc + 4 * c),
                to_lds(&ne[n][4 * c]),
                /*offset=*/0, /*cpol=*/0);
        }
    }
    wait_async0();         // own wave's transfers complete
    __syncthreads();       // all waves' tiles visible
    const float s0 = s_s0;

    // phase 1b: per-row dot with wn, reading from LDS
    #pragma unroll 1
    for (int p = 0; p < 7; ++p) {
        const int n = p * 16 + g;
        if (n < Nn) {
            const float4 v = *(const float4*)(&ne[n][4 * c]);
            float pd = v.x * wn4.x + v.y * wn4.y + v.z * wn4.z + v.w * wn4.w;
            #pragma unroll
            for (int m = 8; m >= 1; m >>= 1) pd += __shfl_xor(pd, m, 16);
            if (c == 0) {
                const float t = s0 + pd;
                tv[n] = t > 0.f ? t : 0.f;           // relu
            }
        }
    }
    __syncthreads();
#else
    __syncthreads();
    const float s0 = s_s0;

    // fallback: stage tile via VGPRs + fused per-row dot
    #pragma unroll 1
    for (int p = 0; p < 7; ++p) {
        const int n = p * 16 + g;
        if (n + 16 < Nn)
            __builtin_prefetch(neb + (size_t)(n + 16) * Hc + 4 * c, 0, 0);
        if (n < Nn) {
            float4 v = *(const float4*)(neb + (size_t)n * Hc + 4 * c);
            *(float4*)(&ne[n][4 * c]) = v;
            float pd = v.x * wn4.x + v.y * wn4.y + v.z * wn4.z + v.w * wn4.w;
            #pragma unroll
            for (int m = 8; m >= 1; m >>= 1) pd += __shfl_xor(pd, m, 16);
            if (c == 0) {
                const float t = s0 + pd;
                tv[n] = t > 0.f ? t : 0.f;           // relu
            }
        }
    }
    __syncthreads();
#endif

    // phase 2: softmax over the 100 scores (wave 0)
    if (tid < 32) {
        float mx = 0.f;                               // relu output >= 0
        for (int n = tid; n < Nn; n += 32) mx = fmaxf(mx, tv[n]);
        #pragma unroll
        for (int m = 16; m >= 1; m >>= 1) mx = fmaxf(mx, __shfl_xor(mx, m, 32));
        float ss = 0.f;
        for (int n = tid; n < Nn; n += 32) {
            const float e = __expf(tv[n] - mx);
            tv[n] = e;
            ss += e;
        }
        #pragma unroll
        for (int m = 16; m >= 1; m >>= 1) ss += __shfl_xor(ss, m, 32);
        if (tid == 0) s_inv = 1.f / ss;
    }
    __syncthreads();

    // phase 3: weighted accumulation out of LDS (deterministic order)
    float4 acc = {0.f, 0.f, 0.f, 0.f};
    for (int n = g; n < Nn; n += 16) {
        const float w = tv[n];
        const float4 v = *(const float4*)(&ne[n][4 * c]);
        acc.x += w * v.x; acc.y += w * v.y; acc.z += w * v.z; acc.w += w * v.w;
    }
    *(float4*)(&red[g][4 * c]) = acc;
    __syncthreads();
    if (tid < 16) {
        const float inv = s_inv;
        float4 s = {0.f, 0.f, 0.f, 0.f};
        #pragma unroll
        for (int gg = 0; gg < 16; ++gg) {
            const float4 v = *(const float4*)(&red[gg][4 * tid]);
            s.x += v.x; s.y += v.y; s.z += v.z; s.w += v.w;
        }
        s.x *= inv; s.y *= inv; s.z *= inv; s.w *= inv;
        *(float4*)(agg_raw + (size_t)row * Hc + 4 * tid) = s;
    }
}

// ---------------------------------------------------------------------------
// Kernel 2: head projection GEMM via V_WMMA_F32_16X16X4_F32 (exact fp32).
//   mh[r, n] = sum_h agg_raw[r,h] * Wbig[h,n] + head_b[n]
//   Wbig[h, n] = head_w[(n>>4)*64*16 + h*16 + (n&15)]
// Block = 16-row M tile, 4 waves = four 16-col N tiles. 16 K-steps of 4.
// All waves fully convergent (WMMA requires EXEC all-1s).
// ---------------------------------------------------------------------------
__global__ __launch_bounds__(128) void k2_head_proj(
    const float* __restrict__ agg_raw,  // [R, 64]
    const float* __restrict__ head_w,   // [4, 64, 16]
    const float* __restrict__ head_b,   // [4, 16] (flat 64)
    float*       __restrict__ mh)       // [R, 64]
{
    const int tid  = threadIdx.x;
    const int lane = tid & 31;
    const int wv   = tid >> 5;          // N tile
    const int r0   = blockIdx.x * 16;   // M tile
    const int n0   = wv * 16;
    const int half = lane >> 4;         // 0: K 0,1 ; 1: K 2,3
    const int lid  = lane & 15;

    const int n = n0 + lid;
    const float* bcol = head_w + (size_t)(n >> 4) * 1024 + (n & 15);

    v8f c = {0.f, 0.f, 0.f, 0.f, 0.f, 0.f, 0.f, 0.f};
    #pragma unroll 1
    for (int kk = 0; kk < 16; ++kk) {
        const int kbase = kk * 4 + half * 2;
        // A frag: lane<16 -> (M=lid, K=kbase,kbase+1); lanes 16-31 -> K=kk*4+2,+3
        const float* ap = agg_raw + (size_t)(r0 + lid) * Hc + kbase;
        v2f a; a.x = ap[0]; a.y = ap[1];
        // B frag mirrors A by row: B[k][n]
        v2f b; b.x = bcol[kbase * 16]; b.y = bcol[(kbase + 1) * 16];
        c = __builtin_amdgcn_wmma_f32_16x16x4_f32(
                /*neg_a=*/false, a, /*neg_b=*/false, b,
                /*c_mod=*/(short)0, c, /*reuse_a=*/false, /*reuse_b=*/false);
    }

    // D layout: VGPR v, lanes 0-15 -> (M=v, N=lid); lanes 16-31 -> (M=v+8, N=lid)
    const float bias = head_b[n];
    #pragma unroll
    for (int v = 0; v < 8; ++v) {
        const int m = v + half * 8;
        mh[(size_t)(r0 + m) * Hc + n] = c[v] + bias;
    }
}

// ---------------------------------------------------------------------------
// Kernel 3: per-batch finalize. agg = mean_l(hist + mh); MLP 192->64->1 + sigmoid.
// agg_hist == hist exactly (softmax over constant rows sums to 1).
// ---------------------------------------------------------------------------
__global__ __launch_bounds__(64) void k3_finalize(
    const int*   __restrict__ user_ids,
    const int*   __restrict__ item_ids,
    const int*   __restrict__ history,
    const float* __restrict__ user_table,
    const float* __restrict__ item_table,
    const float* __restrict__ mh,        // [B*L, 64]
    const float* __restrict__ fc1_w,     // [192, 64]
    const float* __restrict__ fc1_b,     // [64]
    const float* __restrict__ out_w,     // [64]
    const float* __restrict__ out_b,     // [1]
    float*       __restrict__ out)       // [B]
{
    __shared__ float comb[192];
    __shared__ float xs[64];
    const int b = blockIdx.x;
    const int h = threadIdx.x;

    float s = 0.f;
    for (int l = 0; l < Lc; ++l) {
        const int idx = history[b * Lc + l];
        s += item_table[(size_t)idx * Hc + h];
        s += mh[(size_t)(b * Lc + l) * Hc + h];
    }
    comb[Hc + h]      = s * (1.0f / Lc);
    comb[h]           = user_table[(size_t)user_ids[b] * Hc + h];
    comb[2 * Hc + h]  = item_table[(size_t)item_ids[b] * Hc + h];
    __syncthreads();

    float acc = fc1_b[h];
    for (int i = 0; i < 192; ++i) acc += comb[i] * fc1_w[i * Hc + h];
    const float x = acc > 0.f ? acc : 0.f;
    xs[h] = x * out_w[h];
    __syncthreads();

    if (h == 0) {
        float y = out_b[0];
        for (int i = 0; i < Hc; ++i) y += xs[i];
        out[b] = 1.f / (1.f + __expf(-y));
    }
}

// ---------------------------------------------------------------------------
extern "C" void kernel_launch(void* const* d_in, const int* in_sizes, int n_in,
                              void* d_out, int out_size, void* d_ws, size_t ws_size,
                              hipStream_t stream) {
    const int*   user_ids     = (const int*)  d_in[0];
    const int*   item_ids     = (const int*)  d_in[1];
    const int*   history      = (const int*)  d_in[2];
    const float* neighbor_emb = (const float*)d_in[3];
    const float* user_table   = (const float*)d_in[4];
    const float* item_table   = (const float*)d_in[5];
    // d_in[6] w_real, d_in[7] b_real: numerically dead (softmax rows sum to 1)
    const float* w_neigh      = (const float*)d_in[8];
    const float* b_neigh      = (const float*)d_in[9];
    const float* head_w       = (const float*)d_in[10];
    const float* head_b       = (const float*)d_in[11];
    const float* fc1_w        = (const float*)d_in[12];
    const float* fc1_b        = (const float*)d_in[13];
    const float* out_w        = (const float*)d_in[14];
    const float* out_b        = (const float*)d_in[15];
    float*       out          = (float*)d_out;

    float* ws0 = (float*)d_ws;             // agg_raw : R*64 f32 (3.28 MB)
    float* ws1 = ws0 + (size_t)Rrows * Hc; // mh      : R*64 f32 (3.28 MB)

    k1_neighbor_attn<<<Rrows, 256, 0, stream>>>(history, neighbor_emb, item_table,
                                                w_neigh, b_neigh, ws0);
    k2_head_proj<<<Rrows / 16, 128, 0, stream>>>(ws0, head_w, head_b, ws1);
    k3_finalize<<<Bc, 64, 0, stream>>>(user_ids, item_ids, history, user_table,
                                       item_table, ws1, fc1_w, fc1_b, out_w, out_b, out);
}